// DifferentiableRoIAlignRotated_18107582120058
// MI455X (gfx1250) — compile-verified
//
#include <hip/hip_runtime.h>

typedef float v2f __attribute__((ext_vector_type(2)));
typedef float v8f __attribute__((ext_vector_type(8)));

#define OUT_H 7
#define OUT_W 7
#define NPTS 49
#define FC 256
#define FH 128
#define FW 128
#define SCALE 0.25f
#define ROIS_PER_BLK 8
#define CSPLIT 8
#define CPER (FC / CSPLIT)

__global__ __launch_bounds__(256) void roi_align_rotated_kernel(
    const float* __restrict__ features,
    const float* __restrict__ rois,
    float* __restrict__ out)
{
    __shared__ float     s_coord[ROIS_PER_BLK][NPTS][2];
    __shared__ int       s_off[ROIS_PER_BLK][NPTS][4];
    __shared__ float     s_wt [ROIS_PER_BLK][NPTS][4];
    __shared__ long long s_fb [ROIS_PER_BLK];

    const int tid = threadIdx.x;
    const int roiBase = blockIdx.x * ROIS_PER_BLK;

    // ---- Phase 1 (wave 0 only): sampling coords via fp32 WMMA --------------
    // D[p, 2r+j] = gx_p*(w cos) + gy_p*(-h sin) + cx      (j==0 -> xs)
    //            = gx_p*(w sin) + gy_p*( h cos) + cy      (j==1 -> ys)
    if (tid < 32) {
        const int lane = tid;
        const int n    = lane & 15;     // column in B / D
        const int r    = n >> 1;        // roi within group (8 rois x 2 cols)
        const int j    = n & 1;         // 0 = xs, 1 = ys
        const bool hi  = lane >= 16;

        const float* rp = rois + (size_t)(roiBase + r) * 6;
        const float cx = rp[1] * SCALE, cy = rp[2] * SCALE;
        const float w  = rp[3] * SCALE, h  = rp[4] * SCALE;
        const float th = rp[5] * SCALE;            // reference scales theta too
        const float ct = cosf(th), st = sinf(th);  // CLOCKWISE: t = +theta

        // B 4x16 fp32: VGPR0 = {lanes 0-15: K=0, lanes 16-31: K=2},
        //              VGPR1 = {lanes 0-15: K=1, lanes 16-31: K=3}
        v2f B;
        if (!hi) {
            B.x = (j == 0) ? (w * ct) : (w * st);   // K=0 row: coeff of gx
            B.y = (j == 0) ? (-h * st) : (h * ct);  // K=1 row: coeff of gy
        } else {
            B.x = (j == 0) ? cx : cy;               // K=2 row: constant term
            B.y = 0.0f;                             // K=3 row: unused
        }

        for (int tile = 0; tile < 4; ++tile) {
            int m = lane & 15;
            int p = tile * 16 + m;
            if (p > NPTS - 1) p = NPTS - 1;         // clamp pad rows (unstored)
            const float gx = ((p % OUT_W) + 0.5f) / (float)OUT_W - 0.5f;
            const float gy = ((p / OUT_W) + 0.5f) / (float)OUT_H - 0.5f;

            // A 16x4 fp32: lanes 0-15 hold (K=0,K=1)=(gx,gy),
            //              lanes 16-31 hold (K=2,K=3)=(1,0)
            v2f A;
            if (!hi) { A.x = gx;   A.y = gy;   }
            else     { A.x = 1.0f; A.y = 0.0f; }

            v8f D = {};
            D = __builtin_amdgcn_wmma_f32_16x16x4_f32(
                    false, A, false, B, (short)0, D, false, false);

            #pragma unroll
            for (int v = 0; v < 8; ++v) {
                const int pp = tile * 16 + v + (hi ? 8 : 0);
                if (pp < NPTS) s_coord[r][pp][j] = D[v];
            }
        }
    }
    if (tid < ROIS_PER_BLK) {
        const int b = (int)rois[(size_t)(roiBase + tid) * 6 + 0];
        s_fb[tid] = (long long)b * FC * FH * FW;
    }
    __syncthreads();

    // ---- Phase 2: per-(roi,point) taps: 4 offsets + validity-folded weights
    for (int t = tid; t < ROIS_PER_BLK * NPTS; t += 256) {
        const int r = t / NPTS, p = t - r * NPTS;
        const float x = s_coord[r][p][0] - 0.5f;
        const float y = s_coord[r][p][1] - 0.5f;
        const float x0f = floorf(x), y0f = floorf(y);
        const float wx = x - x0f, wy = y - y0f;
        const int x0 = (int)x0f, y0 = (int)y0f;
        const int   txs[2] = { x0, x0 + 1 };
        const int   tys[2] = { y0, y0 + 1 };
        const float wxs[2] = { 1.0f - wx, wx };
        const float wys[2] = { 1.0f - wy, wy };
        #pragma unroll
        for (int ty = 0; ty < 2; ++ty) {
            #pragma unroll
            for (int tx = 0; tx < 2; ++tx) {
                const int xi = txs[tx], yi = tys[ty];
                const bool valid = (xi >= 0) && (xi < FW) && (yi >= 0) && (yi < FH);
                const int xc = xi < 0 ? 0 : (xi > FW - 1 ? FW - 1 : xi);
                const int yc = yi < 0 ? 0 : (yi > FH - 1 ? FH - 1 : yi);
                const int idx = ty * 2 + tx;
                s_off[r][p][idx] = yc * FW + xc;
                s_wt [r][p][idx] = valid ? (wys[ty] * wxs[tx]) : 0.0f;
            }
        }
    }
    __syncthreads();

    // ---- Phase 3: L2-resident gathers + interpolation, coalesced stores ----
    const int cBase = blockIdx.y * CPER;
    for (int r = 0; r < ROIS_PER_BLK; ++r) {
        const long long fb   = s_fb[r];
        const size_t obase   = ((size_t)(roiBase + r) * FC + cBase) * NPTS;
        for (int i = tid; i < CPER * NPTS; i += 256) {
            const int c = i / NPTS, p = i - c * NPTS;
            const float* f = features + fb + (size_t)(cBase + c) * (FH * FW);
            const float v = f[s_off[r][p][0]] * s_wt[r][p][0]
                          + f[s_off[r][p][1]] * s_wt[r][p][1]
                          + f[s_off[r][p][2]] * s_wt[r][p][2]
                          + f[s_off[r][p][3]] * s_wt[r][p][3];
            out[obase + (size_t)c * NPTS + p] = v;
        }
    }
}

extern "C" void kernel_launch(void* const* d_in, const int* in_sizes, int n_in,
                              void* d_out, int out_size, void* d_ws, size_t ws_size,
                              hipStream_t stream) {
    const float* features = (const float*)d_in[0];
    const float* rois     = (const float*)d_in[1];
    float* out            = (float*)d_out;
    const int K = in_sizes[1] / 6;           // 2048 rois
    dim3 grid(K / ROIS_PER_BLK, CSPLIT);     // (256, 8) = 2048 blocks
    roi_align_rotated_kernel<<<grid, 256, 0, stream>>>(features, rois, out);
}